// MMSiSLoss_5583457485021
// MI455X (gfx1250) — compile-verified
//
#include <hip/hip_runtime.h>
#include <stdint.h>

#define H_ 1024
#define W_ 1024
#define C_ 8
#define N_ 2
#define TS 16
#define TCOLS 48
#define PAD_ 16
#define SST 4
#define STEPS 9
#define STEP_ELEMS (C_ * TS * TCOLS) /* 6144 floats = 24KB per step buffer */
#define BLK 256
#define PER_THREAD (STEP_ELEMS / BLK) /* 24 async loads per thread per step */
#define NPIX ((size_t)N_ * H_ * W_)   /* 2,097,152 */
#define NBLK ((int)(NPIX / BLK))      /* 8192 */
#define TILES (N_ * (H_ / TS) * (W_ / TS)) /* 8192 */

// Single-reflection index (halo 16 << 1024, so one reflection suffices).
__device__ __forceinline__ int reflect_idx(int x, int n) {
  x = (x < 0) ? -x : x;
  x = (x >= n) ? (2 * (n - 1) - x) : x;
  return x;
}

// CDNA5 async global->LDS DMA (ASYNCcnt-tracked). LDS dest address is a
// per-lane VGPR byte offset; global address is a per-lane VGPR pair.
__device__ __forceinline__ void async_ld_f32(const float* gaddr, unsigned lds_off) {
  asm volatile("global_load_async_to_lds_b32 %0, %1, off"
               :
               : "v"(lds_off), "v"(gaddr)
               : "memory");
}

// Issue one step's 8ch x 16row x 48col target slab into LDS (async).
__device__ __forceinline__ void issue_step(const float* __restrict__ tgt_n,
                                           int h0, int w0, int tid, int step,
                                           unsigned lds_dst_base) {
#pragma unroll
  for (int k = 0; k < PER_THREAD; ++k) {
    const int e = k * BLK + tid;              // element in step buffer
    const int c = e / (TS * TCOLS);           // channel
    const int rc = e - c * (TS * TCOLS);
    const int r = rc / TCOLS;                 // tile row
    const int col = rc - r * TCOLS;           // tile col (0..47)
    const int grow = reflect_idx(h0 + r + SST * step - PAD_, H_);
    const int gcol = reflect_idx(w0 + col - PAD_, W_);
    const float* g = tgt_n + (size_t)c * (size_t)(H_ * W_) + (size_t)grow * W_ + gcol;
    async_ld_f32(g, lds_dst_base + (unsigned)e * 4u);
  }
}

// ---------------------------------------------------------------------------
// Kernel 1: min over 81 shifts of sum_c |ms - shifted(target)|, per pixel.
// Double-buffered async LDS staging of the per-row-shift target slab.
// ---------------------------------------------------------------------------
__global__ __launch_bounds__(BLK) void k_shift_min(const float* __restrict__ ms_img,
                                                   const float* __restrict__ target,
                                                   float* __restrict__ out_min) {
  __shared__ float tbuf[2][STEP_ELEMS]; // 48KB double buffer
  const int tid = threadIdx.x;
  const int bid = blockIdx.x;
  const int n = bid >> 12;        // 4096 tiles per image (64x64)
  const int rem = bid & 4095;
  const int h0 = (rem >> 6) * TS;
  const int w0 = (rem & 63) * TS;
  const int ty = tid >> 4, tx = tid & 15;
  const int h = h0 + ty, w = w0 + tx;

  const size_t HW = (size_t)H_ * W_;

  // Cache this pixel's 8 channel values of ms_img in registers.
  float msv[C_];
  {
    const float* p = ms_img + (size_t)n * C_ * HW + (size_t)h * W_ + w;
#pragma unroll
    for (int c = 0; c < C_; ++c) msv[c] = p[c * HW];
  }

  const float* tgt_n = target + (size_t)n * C_ * HW;
  const unsigned lds_base = (unsigned)(uintptr_t)(&tbuf[0][0]);

  // Prologue: start DMA of step 0 into buffer 0.
  issue_step(tgt_n, h0, w0, tid, 0, lds_base);

  float m = __builtin_inff();
  for (int i = 0; i < STEPS; ++i) {
    __syncthreads(); // all waves done computing on buf[(i+1)&1] (step i-1)
    if (i + 1 < STEPS) {
      // Prefetch next step while this step is consumed.
      issue_step(tgt_n, h0, w0, tid, i + 1,
                 lds_base + (unsigned)(((i + 1) & 1) * STEP_ELEMS) * 4u);
      // Async loads complete in order: <=24 outstanding => step i landed.
      asm volatile("s_wait_asynccnt 24" ::: "memory");
    } else {
      asm volatile("s_wait_asynccnt 0" ::: "memory");
    }
    __syncthreads(); // all waves' step-i DMA visible in LDS

    const float* base = &tbuf[i & 1][ty * TCOLS + tx];
#pragma unroll
    for (int j = 0; j < STEPS; ++j) {
      float s = 0.0f;
#pragma unroll
      for (int c = 0; c < C_; ++c)
        s += fabsf(msv[c] - base[c * (TS * TCOLS) + j * SST]);
      m = fminf(m, s);
    }
  }
  out_min[(size_t)n * HW + (size_t)h * W_ + w] = m;
}

// ---------------------------------------------------------------------------
// Kernel 2: diff = |mean_c(ms_img_o) - fr_pan|, plus per-block min/max.
// ---------------------------------------------------------------------------
__global__ __launch_bounds__(BLK) void k_diff_minmax(const float* __restrict__ ms_o,
                                                     const float* __restrict__ pan,
                                                     float* __restrict__ diff,
                                                     float* __restrict__ pmin,
                                                     float* __restrict__ pmax) {
  __shared__ float smin[BLK], smax[BLK];
  const size_t HW = (size_t)H_ * W_;
  const int tid = threadIdx.x;
  const size_t pix = (size_t)blockIdx.x * BLK + tid;
  const int n = (int)(pix / HW);
  const size_t hw = pix - (size_t)n * HW;
  const float* p = ms_o + (size_t)n * C_ * HW + hw;
  float s = 0.0f;
#pragma unroll
  for (int c = 0; c < C_; ++c) s += p[c * HW];
  const float d = fabsf(s * 0.125f - pan[pix]);
  diff[pix] = d;
  smin[tid] = d;
  smax[tid] = d;
  __syncthreads();
  for (int off = BLK / 2; off > 0; off >>= 1) {
    if (tid < off) {
      smin[tid] = fminf(smin[tid], smin[tid + off]);
      smax[tid] = fmaxf(smax[tid], smax[tid + off]);
    }
    __syncthreads();
  }
  if (tid == 0) {
    pmin[blockIdx.x] = smin[0];
    pmax[blockIdx.x] = smax[0];
  }
}

// ---------------------------------------------------------------------------
// Kernel 3: final global min/max over block partials (deterministic).
// ---------------------------------------------------------------------------
__global__ __launch_bounds__(BLK) void k_minmax_final(const float* __restrict__ pmin,
                                                      const float* __restrict__ pmax,
                                                      float* __restrict__ lohi) {
  __shared__ float smin[BLK], smax[BLK];
  const int tid = threadIdx.x;
  float lo = __builtin_inff(), hi = -__builtin_inff();
  for (int k = tid; k < NBLK; k += BLK) {
    lo = fminf(lo, pmin[k]);
    hi = fmaxf(hi, pmax[k]);
  }
  smin[tid] = lo;
  smax[tid] = hi;
  __syncthreads();
  for (int off = BLK / 2; off > 0; off >>= 1) {
    if (tid < off) {
      smin[tid] = fminf(smin[tid], smin[tid + off]);
      smax[tid] = fmaxf(smax[tid], smax[tid + off]);
    }
    __syncthreads();
  }
  if (tid == 0) {
    lohi[0] = smin[0];
    lohi[1] = smax[0];
  }
}

// ---------------------------------------------------------------------------
// Kernel 4: mask = normalize(diff)*255 > 10; per-block masked sum + count.
// ---------------------------------------------------------------------------
__global__ __launch_bounds__(BLK) void k_masked_partial(const float* __restrict__ minsl,
                                                        const float* __restrict__ diff,
                                                        const float* __restrict__ lohi,
                                                        float* __restrict__ psum,
                                                        float* __restrict__ pcnt) {
  __shared__ float ssum[BLK], scnt[BLK];
  const int tid = threadIdx.x;
  const float lo = lohi[0], hi = lohi[1];
  const size_t pix = (size_t)blockIdx.x * BLK + tid;
  const float d = diff[pix];
  // hi==lo -> 0/0 = NaN -> compare false -> mask empty (matches reference).
  const float norm = (d - lo) / (hi - lo) * 255.0f;
  const bool mk = norm > 10.0f;
  ssum[tid] = mk ? minsl[pix] : 0.0f;
  scnt[tid] = mk ? 1.0f : 0.0f;
  __syncthreads();
  for (int off = BLK / 2; off > 0; off >>= 1) {
    if (tid < off) {
      ssum[tid] += ssum[tid + off];
      scnt[tid] += scnt[tid + off];
    }
    __syncthreads();
  }
  if (tid == 0) {
    psum[blockIdx.x] = ssum[0];
    pcnt[blockIdx.x] = scnt[0];
  }
}

// ---------------------------------------------------------------------------
// Kernel 5: final scalar = cnt>0 ? sum/max(cnt,1) : 0.
// ---------------------------------------------------------------------------
__global__ __launch_bounds__(BLK) void k_final(const float* __restrict__ psum,
                                               const float* __restrict__ pcnt,
                                               float* __restrict__ out) {
  __shared__ float ssum[BLK], scnt[BLK];
  const int tid = threadIdx.x;
  float s = 0.0f, c = 0.0f;
  for (int k = tid; k < NBLK; k += BLK) {
    s += psum[k];
    c += pcnt[k];
  }
  ssum[tid] = s;
  scnt[tid] = c;
  __syncthreads();
  for (int off = BLK / 2; off > 0; off >>= 1) {
    if (tid < off) {
      ssum[tid] += ssum[tid + off];
      scnt[tid] += scnt[tid + off];
    }
    __syncthreads();
  }
  if (tid == 0) {
    const float cnt = scnt[0];
    out[0] = (cnt > 0.0f) ? (ssum[0] / fmaxf(cnt, 1.0f)) : 0.0f;
  }
}

extern "C" void kernel_launch(void* const* d_in, const int* in_sizes, int n_in,
                              void* d_out, int out_size, void* d_ws, size_t ws_size,
                              hipStream_t stream) {
  (void)in_sizes; (void)n_in; (void)out_size; (void)ws_size;
  const float* ms_img = (const float*)d_in[0];
  const float* target = (const float*)d_in[1];
  const float* ms_o   = (const float*)d_in[2];
  const float* pan    = (const float*)d_in[3];
  float* out = (float*)d_out;

  // Workspace layout (floats): minsl[NPIX] | diff[NPIX] | pmin[NBLK] |
  // pmax[NBLK] | lohi[8] | psum[NBLK] | pcnt[NBLK]   (~16.9 MB)
  float* ws   = (float*)d_ws;
  float* A    = ws;              // min shift loss per pixel
  float* B    = A + NPIX;        // |gray - pan| per pixel
  float* pmin = B + NPIX;
  float* pmax = pmin + NBLK;
  float* lohi = pmax + NBLK;
  float* psum = lohi + 8;
  float* pcnt = psum + NBLK;

  k_shift_min<<<TILES, BLK, 0, stream>>>(ms_img, target, A);
  k_diff_minmax<<<NBLK, BLK, 0, stream>>>(ms_o, pan, B, pmin, pmax);
  k_minmax_final<<<1, BLK, 0, stream>>>(pmin, pmax, lohi);
  k_masked_partial<<<NBLK, BLK, 0, stream>>>(A, B, lohi, psum, pcnt);
  k_final<<<1, BLK, 0, stream>>>(psum, pcnt, out);
}